// LearnableDriftCompensation_34806414967387
// MI455X (gfx1250) — compile-verified
//
#include <hip/hip_runtime.h>
#include <hip/hip_bf16.h>
#include <math.h>

typedef float v2f __attribute__((ext_vector_type(2)));
typedef float v4f __attribute__((ext_vector_type(4)));
typedef float v8f __attribute__((ext_vector_type(8)));

#define TWO_PI_F 6.283185307179586f

// ---------------------------------------------------------------------------
// K1: per-class prep. One block per class c.
//   out_comp[c,:] = prototypes[c,:]            (drift kernel overwrites later)
//   invvar[c,:]   = 1/(softplus(gvars)+1e-4)
//   logvar[c]     = sum_d clip(log(2*pi*var), -10, 10)
//   sums[c] = counts[c] = 0
// ---------------------------------------------------------------------------
__global__ __launch_bounds__(256) void prep_kernel(
    const float* __restrict__ proto, const float* __restrict__ gvars,
    float* __restrict__ out_comp, float* __restrict__ invvar,
    float* __restrict__ logvar, float* __restrict__ sums,
    float* __restrict__ counts, int D)
{
  const int c = blockIdx.x;
  float lt = 0.0f;
  for (int d = threadIdx.x; d < D; d += 256) {
    size_t idx = (size_t)c * D + d;
    out_comp[idx] = proto[idx];
    float x  = gvars[idx];
    float sp = (x > 15.0f) ? x : __logf(1.0f + __expf(x));   // softplus
    float v  = sp + 1e-4f;
    invvar[idx] = 1.0f / v;
    lt += fminf(fmaxf(__logf(TWO_PI_F * v), -10.0f), 10.0f);
  }
  __shared__ float red[256];
  red[threadIdx.x] = lt;
  __syncthreads();
  for (int s = 128; s > 0; s >>= 1) {
    if ((int)threadIdx.x < s) red[threadIdx.x] += red[threadIdx.x + s];
    __syncthreads();
  }
  if (threadIdx.x == 0) {
    logvar[c] = red[0];
    sums[c]   = 0.0f;
    counts[c] = 0.0f;
  }
}

// ---------------------------------------------------------------------------
// K2: drift compensation on selected rows. One block per class_indices entry.
// Reads only inputs -> order independent even with duplicate indices.
// ---------------------------------------------------------------------------
__global__ __launch_bounds__(256) void drift_kernel(
    const float* __restrict__ proto, const float* __restrict__ driftc,
    const float* __restrict__ momentum, const int* __restrict__ cidx,
    float* __restrict__ out_comp, int D)
{
  const int ci   = cidx[blockIdx.x];
  const float mg = 1.0f / (1.0f + __expf(-momentum[ci]));   // sigmoid
  const float sc = 1.0f - mg;
  for (int d = threadIdx.x; d < D; d += 256) {
    size_t idx = (size_t)ci * D + d;
    out_comp[idx] = proto[idx] + sc * driftc[idx];
  }
}

// ---------------------------------------------------------------------------
// K3: streaming dsq kernel. 128 threads = 4 waves; each wave owns 16 samples
// (A-matrix M). B = all-ones 4x16, so WMMA sums A's K dimension into every
// accumulator column. Because everything is summed over the whole loop and B
// is ones, the d->(k,step) mapping is an arbitrary permutation: each lane
// loads one contiguous float4 (lanes<16: d=[db,db+4), lanes>=16: d=[db+4,db+8))
// and two WMMAs consume quad elements {0,1} and {2,3}. Full b128 loads,
// two independent accumulator chains.
// ---------------------------------------------------------------------------
__global__ __launch_bounds__(128) void dsq_wmma_kernel(
    const float* __restrict__ features, const float* __restrict__ gmeans,
    const float* __restrict__ invvar, const int* __restrict__ labels,
    float* __restrict__ sums, float* __restrict__ counts, int N, int D)
{
  const int lane      = threadIdx.x & 31;
  const int wave      = threadIdx.x >> 5;
  const int wave_base = (blockIdx.x * 4 + wave) * 16;
  const int m         = lane & 15;
  const int h         = lane >> 4;                 // lane half: 0 or 1

  int sm  = wave_base + m;
  int scl = (sm < N) ? sm : (N - 1);               // clamp tail; rows unused
  int lbl = labels[scl];

  const float* fp = features + (size_t)scl * D + 4 * h;
  const float* mp = gmeans   + (size_t)lbl * D + 4 * h;
  const float* vp = invvar   + (size_t)lbl * D + 4 * h;

  v2f ones; ones.x = 1.0f; ones.y = 1.0f;
  v8f acc0 = {};
  v8f acc1 = {};

  int db = 0;
  for (; db + 7 < D; db += 8) {
    v4f f = *(const v4f*)(fp + db);
    v4f g = *(const v4f*)(mp + db);
    v4f w = *(const v4f*)(vp + db);

    v4f dl = f - g;
    v4f t  = dl * dl * w;
    v2f a0, a1;
    a0.x = fminf(fmaxf(t.x, 0.0f), 100.0f);
    a0.y = fminf(fmaxf(t.y, 0.0f), 100.0f);
    a1.x = fminf(fmaxf(t.z, 0.0f), 100.0f);
    a1.y = fminf(fmaxf(t.w, 0.0f), 100.0f);

    acc0 = __builtin_amdgcn_wmma_f32_16x16x4_f32(
        false, a0, false, ones, (short)0, acc0, false, false);
    acc1 = __builtin_amdgcn_wmma_f32_16x16x4_f32(
        false, a1, false, ones, (short)0, acc1, false, false);
  }
  for (; db + 3 < D; db += 4) {                    // D % 8 == 4 remainder
    const int kp = 2 * h;                          // K-pair layout for one step
    v2f f0 = *(const v2f*)(features + (size_t)scl * D + db + kp);
    v2f g0 = *(const v2f*)(gmeans   + (size_t)lbl * D + db + kp);
    v2f w0 = *(const v2f*)(invvar   + (size_t)lbl * D + db + kp);
    float d00 = f0.x - g0.x, d01 = f0.y - g0.y;
    v2f a0;
    a0.x = fminf(fmaxf(d00 * d00 * w0.x, 0.0f), 100.0f);
    a0.y = fminf(fmaxf(d01 * d01 * w0.y, 0.0f), 100.0f);
    acc0 = __builtin_amdgcn_wmma_f32_16x16x4_f32(
        false, a0, false, ones, (short)0, acc0, false, false);
  }

  v8f acc = acc0 + acc1;

  // C/D layout: lane 0 (N=0) VGPR i -> sample i; lane 16 VGPR i -> sample 8+i.
  if ((lane & 15) == 0) {
    int base = wave_base + h * 8;
#pragma unroll
    for (int i = 0; i < 8; ++i) {
      int s = base + i;
      if (s < N) {
        int l = labels[s];
        atomicAdd(&sums[l],   acc[i]);
        atomicAdd(&counts[l], 1.0f);
      }
    }
  }
}

// ---------------------------------------------------------------------------
// K4: finalize loss (single block).
// ---------------------------------------------------------------------------
__global__ __launch_bounds__(256) void finalize_kernel(
    const float* __restrict__ sums, const float* __restrict__ counts,
    const float* __restrict__ logvar, float* __restrict__ loss_out, int C)
{
  float nll_sum = 0.0f, np = 0.0f;
  for (int c = threadIdx.x; c < C; c += 256) {
    float cnt  = counts[c];
    float mdsq = sums[c] / fmaxf(cnt, 1.0f);
    float nll  = fminf(fmaxf(0.5f * (logvar[c] + mdsq), 0.0f), 10.0f);
    if (cnt > 0.0f) { nll_sum += nll; np += 1.0f; }
  }
  __shared__ float r0[256];
  __shared__ float r1[256];
  r0[threadIdx.x] = nll_sum;
  r1[threadIdx.x] = np;
  __syncthreads();
  for (int s = 128; s > 0; s >>= 1) {
    if ((int)threadIdx.x < s) {
      r0[threadIdx.x] += r0[threadIdx.x + s];
      r1[threadIdx.x] += r1[threadIdx.x + s];
    }
    __syncthreads();
  }
  if (threadIdx.x == 0) loss_out[0] = r0[0] / fmaxf(r1[0], 1.0f);
}

// ---------------------------------------------------------------------------
extern "C" void kernel_launch(void* const* d_in, const int* in_sizes, int n_in,
                              void* d_out, int out_size, void* d_ws, size_t ws_size,
                              hipStream_t stream) {
  const float* features  = (const float*)d_in[0];
  const float* proto     = (const float*)d_in[1];
  const float* driftc    = (const float*)d_in[2];
  const float* momentum  = (const float*)d_in[3];
  const float* gmeans    = (const float*)d_in[4];
  const float* gvars     = (const float*)d_in[5];
  const int*   labels    = (const int*)d_in[6];
  const int*   cidx      = (const int*)d_in[7];

  const int C = in_sizes[3];                 // drift_momentum is [C]
  const int N = in_sizes[6];                 // labels is [N]
  const int D = in_sizes[0] / N;             // features is [N, D]
  const int K = in_sizes[7];                 // class_indices length

  float* out_comp = (float*)d_out;           // [C*D]
  float* out_loss = out_comp + (size_t)C * D;

  // workspace layout: invvar[C*D] | logvar[C] | sums[C] | counts[C]
  float* invvar = (float*)d_ws;
  float* logvar = invvar + (size_t)C * D;
  float* sums   = logvar + C;
  float* counts = sums + C;

  prep_kernel<<<C, 256, 0, stream>>>(proto, gvars, out_comp, invvar,
                                     logvar, sums, counts, D);
  drift_kernel<<<K, 256, 0, stream>>>(proto, driftc, momentum, cidx,
                                      out_comp, D);
  const int samples_per_block = 64;          // 4 waves x 16 samples
  const int blocks = (N + samples_per_block - 1) / samples_per_block;
  dsq_wmma_kernel<<<blocks, 128, 0, stream>>>(features, gmeans, invvar,
                                              labels, sums, counts, N, D);
  finalize_kernel<<<1, 256, 0, stream>>>(sums, counts, logvar, out_loss, C);
}